// nt_xent_65618510349014
// MI455X (gfx1250) — compile-verified
//
#include <hip/hip_runtime.h>
#include <math.h>

#define N_TOTAL 8192
#define HALF_N  4096
#define DIMK    128
#define XMAX    2.0f      // logits bounded: |sim|<=1, /T=*2 -> [-2,2]
#define TEMP_INV 2.0f     // 1 / 0.5
#define NSPLIT  4         // column-sweep split per row tile (waves per block)
#define CT_PER_WAVE ((N_TOTAL / 16) / NSPLIT)   // 128 column tiles per wave

typedef __attribute__((ext_vector_type(2))) float v2f;
typedef __attribute__((ext_vector_type(4))) float v4f;
typedef __attribute__((ext_vector_type(8))) float v8f;

// ---------------------------------------------------------------------------
// Kernel 1: row-wise L2 normalize [z1 ; z2] -> zn (8192 x 128 f32)
// One wave32 per row: each lane owns 4 contiguous floats (128 = 32*4).
// ---------------------------------------------------------------------------
__global__ void __launch_bounds__(256) nk_normalize(const float* __restrict__ z1,
                                                    const float* __restrict__ z2,
                                                    float* __restrict__ zn) {
    const int lane = threadIdx.x & 31;
    const int wave = threadIdx.x >> 5;
    const int row  = blockIdx.x * 8 + wave;

    const float* src = (row < HALF_N) ? (z1 + (size_t)row * DIMK)
                                      : (z2 + (size_t)(row - HALF_N) * DIMK);
    v4f v = *(const v4f*)(src + lane * 4);
    float ss = v.x * v.x + v.y * v.y + v.z * v.z + v.w * v.w;
#pragma unroll
    for (int m = 16; m >= 1; m >>= 1)
        ss += __shfl_xor(ss, m, 32);
    const float inv = 1.0f / fmaxf(sqrtf(ss), 1e-12f);
    v.x *= inv; v.y *= inv; v.z *= inv; v.w *= inv;
    *(v4f*)(zn + (size_t)row * DIMK + lane * 4) = v;
}

// ---------------------------------------------------------------------------
// Kernel 2: fused sim-GEMM + masked exp-sum + positive-pair gather.
// Block b, wave w: row tile b (16 rows, A fragment resident = 64 VGPRs),
// column tiles [w*128, w*128+128) swept with V_WMMA_F32_16X16X4_F32 chains.
// Since logits are bounded by XMAX, per-row state is a plain sum of
// exp(x - XMAX) -> additive across the 4 column splits (combined later).
// f32 WMMA A(16x4)/B(4x16) per-lane layout for Z*Z^T is the same gather:
//   lane L<16 : row/col base+L, K = {k0, k0+1}
//   lane L>=16: row/col base+(L-16), K = {k0+2, k0+3}
// C layout: VGPR v, lane L -> (M = v + 8*(L>=16), N = L&15).
// ---------------------------------------------------------------------------
__global__ void __launch_bounds__(128) nk_simloss(const float* __restrict__ zn,
                                                  float* __restrict__ sBuf,
                                                  float* __restrict__ tBuf) {
    const int lane    = threadIdx.x & 31;
    const int wave    = threadIdx.x >> 5;        // column split 0..3
    const int tileRow = blockIdx.x;              // 512 row tiles
    const int rowBase = tileRow * 16;
    const int ct0     = wave * CT_PER_WAVE;
    const int half    = lane >> 4;               // 0 | 1
    const int l15     = lane & 15;
    const int koff    = half * 2;                // K sub-offset inside 4-chunk

    // Resident A fragment: 32 chunks of float2 (K = 128 in steps of 4)
    v2f a[32];
    {
        const float* ap = zn + (size_t)(rowBase + l15) * DIMK + koff;
#pragma unroll
        for (int kc = 0; kc < 32; ++kc)
            a[kc] = *(const v2f*)(ap + kc * 4);
    }

    float s[8];                 // sum of exp(x - XMAX) over this lane's columns
    float t[8];                 // positive-pair logit (exactly one lane/wave sees it)
#pragma unroll
    for (int v = 0; v < 8; ++v) { s[v] = 0.0f; t[v] = -3.0e38f; }

    for (int ct = ct0; ct < ct0 + CT_PER_WAVE; ++ct) {
        const int colBase = ct * 16;

        v2f b[32];
        const float* bp = zn + (size_t)(colBase + l15) * DIMK + koff;
#pragma unroll
        for (int kc = 0; kc < 32; ++kc)
            b[kc] = *(const v2f*)(bp + kc * 4);

        v8f acc = {};
#pragma unroll
        for (int kc = 0; kc < 32; ++kc)
            acc = __builtin_amdgcn_wmma_f32_16x16x4_f32(
                /*neg_a=*/false, a[kc], /*neg_b=*/false, b[kc],
                /*c_mod=*/(short)0, acc, /*reuse_a=*/false, /*reuse_b=*/false);

        const int gcol = colBase + l15;
#pragma unroll
        for (int v = 0; v < 8; ++v) {
            const int grow = rowBase + v + half * 8;
            const int gtgt = (grow + HALF_N) & (N_TOTAL - 1);
            const float x  = acc[v] * TEMP_INV;
            s[v] += (gcol != grow) ? __expf(x - XMAX) : 0.0f;  // diagonal = -inf
            t[v]  = (gcol == gtgt) ? x : t[v];
        }
    }

    // Reduce across the 16 lanes of each half-wave (masks 1,2,4,8 keep bit 4)
#pragma unroll
    for (int m = 1; m <= 8; m <<= 1) {
#pragma unroll
        for (int v = 0; v < 8; ++v) {
            s[v] += __shfl_xor(s[v], m, 32);
            t[v]  = fmaxf(t[v], __shfl_xor(t[v], m, 32));
        }
    }

    // Per-wave partials: 16 rows. Rows v+0..7 live in lanes 0..15 (lane 0
    // writes), rows 8..15 in lanes 16..31 (lane 16 writes).
    if (l15 == 0) {
        const size_t base = ((size_t)tileRow * NSPLIT + wave) * 16 + half * 8;
#pragma unroll
        for (int v = 0; v < 8; ++v) {
            sBuf[base + v] = s[v];
            tBuf[base + v] = t[v];
        }
    }
}

// ---------------------------------------------------------------------------
// Kernel 3: deterministic finalize. For each of 8192 rows combine the 4
// column-split partials (sum s, max t), form lse - t, tree-reduce, average.
// ---------------------------------------------------------------------------
__global__ void __launch_bounds__(1024) nk_finalize(const float* __restrict__ sBuf,
                                                    const float* __restrict__ tBuf,
                                                    float* __restrict__ out) {
    __shared__ float sm[1024];
    const int tid = threadIdx.x;

    float local = 0.0f;
#pragma unroll
    for (int i = 0; i < N_TOTAL / 1024; ++i) {
        const int r        = tid + i * 1024;
        const int tileRow  = r >> 4;
        const int rowInTile= r & 15;
        float ssum = 0.0f;
        float tmax = -3.0e38f;
#pragma unroll
        for (int q = 0; q < NSPLIT; ++q) {
            const size_t idx = ((size_t)tileRow * NSPLIT + q) * 16 + rowInTile;
            ssum += sBuf[idx];
            tmax  = fmaxf(tmax, tBuf[idx]);
        }
        local += (XMAX + __logf(ssum)) - tmax;   // logsumexp_r - target_r
    }

    sm[tid] = local;
    __syncthreads();
#pragma unroll
    for (int stride = 512; stride > 0; stride >>= 1) {
        if (tid < stride) sm[tid] += sm[tid + stride];
        __syncthreads();
    }
    if (tid == 0)
        out[0] = sm[0] / (float)N_TOTAL;         // loss = mean(lse_i - t_i)
}

// ---------------------------------------------------------------------------
extern "C" void kernel_launch(void* const* d_in, const int* in_sizes, int n_in,
                              void* d_out, int out_size, void* d_ws, size_t ws_size,
                              hipStream_t stream) {
    const float* z1 = (const float*)d_in[0];
    const float* z2 = (const float*)d_in[1];

    char* ws = (char*)d_ws;
    float* zn   = (float*)ws;                                   // 8192*128 f32 = 4 MB
    ws += (size_t)N_TOTAL * DIMK * sizeof(float);
    float* sBuf = (float*)ws;                                   // 512*4*16 f32 = 128 KB
    ws += (size_t)(N_TOTAL / 16) * NSPLIT * 16 * sizeof(float);
    float* tBuf = (float*)ws;                                   // 128 KB

    nk_normalize<<<N_TOTAL / 8, 256, 0, stream>>>(z1, z2, zn);
    nk_simloss  <<<N_TOTAL / 16, 32 * NSPLIT, 0, stream>>>(zn, sBuf, tBuf);
    nk_finalize <<<1, 1024, 0, stream>>>(sBuf, tBuf, (float*)d_out);
}